// SteganographicReasoningDecoder_5033701671606
// MI455X (gfx1250) — compile-verified
//
#include <hip/hip_runtime.h>
#include <hip/hip_bf16.h>
#include <math.h>
#include <stdint.h>

typedef __attribute__((ext_vector_type(2))) float v2f;
typedef __attribute__((ext_vector_type(8))) float v8f;

// Problem constants
constexpr int Bn = 256;    // batch
constexpr int Lf = 5000;   // signal length
constexpr int Cn = 3;      // channels
constexpr int Hh = 256;    // fc1 hidden

// Fused-conv tiling
constexpr int T_TILE = 112;            // env outputs per block
constexpr int H2LEN  = 128;            // padded h2 rows (need T+6=118) -> 8 WMMA m-tiles
constexpr int H1LEN  = 144;            // padded h1 rows (need T+20=132; A reads up to 127+15=142)
constexpr int XLEN   = 176;            // x tile (need T+50=162; conv1 reads up to 143+30=173)
constexpr int NTILE_ROW = 45;          // ceil(5000/112)

// -------------------- Kernel 1: pad conv2 weights 15 -> 16 taps --------------------
__global__ void k_w2pad(const float* __restrict__ w2, float* __restrict__ w2p) {
    int i = blockIdx.x * blockDim.x + threadIdx.x;          // C*64*32*16 elements
    if (i >= Cn * 64 * 32 * 16) return;
    int tap = i & 15;
    int rest = i >> 4;                                      // (c*64+oc)*32+ic
    w2p[i] = (tap < 15) ? w2[rest * 15 + tap] : 0.0f;
}

// -------------------- Kernel 2: SOS IIR filter (direct form II transposed) ---------
__global__ void k_sosfilt(const float* __restrict__ sig, const float* __restrict__ sos,
                          float* __restrict__ filt) {
    int idx = blockIdx.x * blockDim.x + threadIdx.x;        // c*Bn + b
    if (idx >= Cn * Bn) return;
    int c = idx / Bn, b = idx % Bn;

    float b0[4], b1c[4], b2c[4], a1[4], a2[4], s1[4], s2[4];
#pragma unroll
    for (int s = 0; s < 4; s++) {
        const float* p = sos + (c * 4 + s) * 6;
        b0[s] = p[0]; b1c[s] = p[1]; b2c[s] = p[2]; a1[s] = p[4]; a2[s] = p[5];
        s1[s] = 0.0f; s2[s] = 0.0f;
    }
    const float* x = sig + (size_t)b * Lf;
    float* y = filt + (size_t)idx * Lf;
    for (int l = 0; l < Lf; l++) {
        if ((l & 63) == 0) __builtin_prefetch(x + l + 256, 0, 1);
        float v = x[l];
#pragma unroll
        for (int s = 0; s < 4; s++) {
            float yy = fmaf(b0[s], v, s1[s]);
            s1[s] = fmaf(b1c[s], v, fmaf(-a1[s], yy, s2[s]));
            s2[s] = fmaf(b2c[s], v, -a2[s] * yy);
            v = yy;
        }
        y[l] = v;
    }
}

// -------------------- Kernel 3: fused conv1+conv2(WMMA)+conv3 ----------------------
__global__ __launch_bounds__(256)
void k_convs(const float* __restrict__ filt,
             const float* __restrict__ w1, const float* __restrict__ b1,
             const float* __restrict__ b2,
             const float* __restrict__ w3, const float* __restrict__ b3,
             const float* __restrict__ w2p,
             float* __restrict__ env) {
    __shared__ float xs[XLEN];
    __shared__ float h1[32 * H1LEN];
    __shared__ float h2[64 * H2LEN];
    __shared__ float w1s[32 * 31];
    __shared__ float b1s[32];
    __shared__ float b2s[64];
    __shared__ float w3s[64 * 7];
    __shared__ float b3s;

    int tid = threadIdx.x;
    int tilei = blockIdx.x % NTILE_ROW;
    int row   = blockIdx.x / NTILE_ROW;          // c*Bn + b
    int c = row / Bn;
    int t0 = tilei * T_TILE;
    const float* xrow = filt + (size_t)row * Lf;

    // stage small weights in LDS
    for (int i = tid; i < 32 * 31; i += 256) w1s[i] = w1[c * 32 * 31 + i];
    for (int i = tid; i < 32; i += 256) b1s[i] = b1[c * 32 + i];
    for (int i = tid; i < 64; i += 256) b2s[i] = b2[c * 64 + i];
    for (int i = tid; i < 64 * 7; i += 256) w3s[i] = w3[c * 64 * 7 + i];
    if (tid == 0) b3s = b3[c];

    // stage x tile: interior tiles via async global->LDS (ASYNCcnt path),
    // boundary tiles via zero-padded scalar path. Block-uniform branch.
    if (t0 >= 25 && t0 + (XLEN - 25) <= Lf) {
        if (tid < XLEN) {
            const float* g = xrow + (t0 - 25 + tid);
            unsigned ldsoff = (unsigned)(uintptr_t)&xs[tid];   // group-segment offset
            asm volatile("global_load_async_to_lds_b32 %0, %1, off"
                         :: "v"(ldsoff), "v"(g) : "memory");
        }
        asm volatile("s_wait_asynccnt 0" ::: "memory");
    } else {
        for (int q = tid; q < XLEN; q += 256) {
            int ab = t0 - 25 + q;
            xs[q] = (ab >= 0 && ab < Lf) ? xrow[ab] : 0.0f;
        }
    }
    __syncthreads();

    // conv1: 32ch, k=31, relu; h1 index i <-> abs time t0-10+i
    for (int o = tid; o < 32 * H1LEN; o += 256) {
        int ic = o / H1LEN;
        int i  = o - ic * H1LEN;
        int ab = t0 - 10 + i;
        float acc = b1s[ic];
#pragma unroll
        for (int k = 0; k < 31; k++) acc = fmaf(w1s[ic * 31 + k], xs[i + k], acc);
        h1[o] = (ab >= 0 && ab < Lf) ? fmaxf(acc, 0.0f) : 0.0f;
    }
    __syncthreads();

    // conv2 as implicit GEMM via V_WMMA_F32_16X16X4_F32.
    // M = h2 row j (abs time t0-3+j), N = out channel, K = ic*16+tap (tap 15 is zero weight).
    int lane = tid & 31, wv = tid >> 5;
    int q = lane & 15, half = lane >> 4;
    for (int tile = wv; tile < 32; tile += 8) {           // 8 m-tiles x 4 n-tiles
        int mt = tile >> 2, nt = tile & 3;
        int n = nt * 16 + q;                              // this lane's column (out channel)
        int j = mt * 16 + q;                              // this lane's A row (time)
        float bias = b2s[n];
        v8f acc;
#pragma unroll
        for (int r = 0; r < 8; r++) acc[r] = bias;
        const float* wrow = w2p + ((size_t)(c * 64 + n)) * 512;
        for (int kk = 0; kk < 512; kk += 4) {
            int k0 = kk + half * 2;                       // lanes 16-31 hold K+2 (ISA f32 A/B layout)
            int k1 = k0 + 1;
            v2f a, bb;
            a.x = h1[(k0 >> 4) * H1LEN + j + (k0 & 15)];
            a.y = h1[(k1 >> 4) * H1LEN + j + (k1 & 15)];
            bb.x = wrow[k0];
            bb.y = wrow[k1];
            acc = __builtin_amdgcn_wmma_f32_16x16x4_f32(false, a, false, bb,
                                                        (short)0, acc, false, false);
        }
#pragma unroll
        for (int r = 0; r < 8; r++) {
            int jr = mt * 16 + r + half * 8;              // D row (lanes 16-31 hold M+8)
            int ab = t0 - 3 + jr;
            float v = fmaxf(acc[r], 0.0f);
            h2[n * H2LEN + jr] = (jr < T_TILE + 6 && ab >= 0 && ab < Lf) ? v : 0.0f;
        }
    }
    __syncthreads();

    // conv3: 64ch -> 1, k=7, sigmoid -> env
    if (tid < T_TILE) {
        int t = tid;
        int ab = t0 + t;
        if (ab < Lf) {
            float acc = b3s;
            for (int ic = 0; ic < 64; ic++) {
#pragma unroll
                for (int k = 0; k < 7; k++)
                    acc = fmaf(w3s[ic * 7 + k], h2[ic * H2LEN + t + k], acc);
            }
            env[(size_t)row * Lf + ab] = 1.0f / (1.0f + __expf(-acc));
        }
    }
}

// -------------------- Kernel 4: fc1 GEMM (M=256,N=256,K=5000) + ReLU ---------------
__global__ __launch_bounds__(256)
void k_fc1(const float* __restrict__ env, const float* __restrict__ fw,
           const float* __restrict__ fb, float* __restrict__ hid) {
    int gwv = (blockIdx.x * blockDim.x + threadIdx.x) >> 5;   // 0..767 wave-tiles
    int lane = threadIdx.x & 31;
    int c = gwv >> 8;
    int rem = gwv & 255;
    int mt = rem >> 4, nt = rem & 15;
    int q = lane & 15, half = lane >> 4;

    const float* A  = env + ((size_t)(c * Bn + mt * 16)) * Lf;   // rows = batch, stride Lf
    const float* Bp = fw  + ((size_t)(c * Hh + nt * 16)) * Lf;   // rows = hidden, stride Lf
    float bias = fb[c * Hh + nt * 16 + q];
    v8f acc;
#pragma unroll
    for (int r = 0; r < 8; r++) acc[r] = bias;

    for (int kk = 0; kk < Lf; kk += 4) {
        int k0 = kk + half * 2;
        v2f a, bb;
        a.x  = A[q * Lf + k0];  a.y  = A[q * Lf + k0 + 1];
        bb.x = Bp[q * Lf + k0]; bb.y = Bp[q * Lf + k0 + 1];
        acc = __builtin_amdgcn_wmma_f32_16x16x4_f32(false, a, false, bb,
                                                    (short)0, acc, false, false);
    }
#pragma unroll
    for (int r = 0; r < 8; r++) {
        int m = mt * 16 + r + half * 8;
        int n = nt * 16 + q;
        hid[((size_t)(c * Bn + m)) * Hh + n] = fmaxf(acc[r], 0.0f);
    }
}

// -------------------- Kernel 5: fc2 GEMM (M=256,N=64,K=256) + sigmoid --------------
__global__ __launch_bounds__(256)
void k_fc2(const float* __restrict__ hid, const float* __restrict__ fw,
           const float* __restrict__ fb, float* __restrict__ out) {
    int gwv = (blockIdx.x * blockDim.x + threadIdx.x) >> 5;   // 0..191 wave-tiles
    int lane = threadIdx.x & 31;
    int c = gwv >> 6;
    int rem = gwv & 63;
    int mt = rem >> 2, nt = rem & 3;
    int q = lane & 15, half = lane >> 4;

    const float* A  = hid + ((size_t)(c * Bn + mt * 16)) * Hh;
    const float* Bp = fw  + ((size_t)(c * 64 + nt * 16)) * Hh;
    float bias = fb[c * 64 + nt * 16 + q];
    v8f acc;
#pragma unroll
    for (int r = 0; r < 8; r++) acc[r] = bias;

    for (int kk = 0; kk < Hh; kk += 4) {
        int k0 = kk + half * 2;
        v2f a, bb;
        a.x  = A[q * Hh + k0];  a.y  = A[q * Hh + k0 + 1];
        bb.x = Bp[q * Hh + k0]; bb.y = Bp[q * Hh + k0 + 1];
        acc = __builtin_amdgcn_wmma_f32_16x16x4_f32(false, a, false, bb,
                                                    (short)0, acc, false, false);
    }
#pragma unroll
    for (int r = 0; r < 8; r++) {
        int m = mt * 16 + r + half * 8;                  // batch index
        int o = nt * 16 + q;                             // output bit index
        out[(size_t)m * (Cn * 64) + c * 64 + o] = 1.0f / (1.0f + __expf(-acc[r]));
    }
}

// -------------------- Launcher -----------------------------------------------------
extern "C" void kernel_launch(void* const* d_in, const int* in_sizes, int n_in,
                              void* d_out, int out_size, void* d_ws, size_t ws_size,
                              hipStream_t stream) {
    (void)in_sizes; (void)n_in; (void)out_size; (void)ws_size;

    const float* signal  = (const float*)d_in[0];
    const float* sos     = (const float*)d_in[1];
    const float* conv1_w = (const float*)d_in[2];
    const float* conv1_b = (const float*)d_in[3];
    const float* conv2_w = (const float*)d_in[4];
    const float* conv2_b = (const float*)d_in[5];
    const float* conv3_w = (const float*)d_in[6];
    const float* conv3_b = (const float*)d_in[7];
    const float* fc1_w   = (const float*)d_in[8];
    const float* fc1_b   = (const float*)d_in[9];
    const float* fc2_w   = (const float*)d_in[10];
    const float* fc2_b   = (const float*)d_in[11];
    float* out = (float*)d_out;

    // workspace layout (floats): filt | env | w2pad | hid   (~32 MB total)
    float* ws   = (float*)d_ws;
    float* filt = ws;
    float* env  = filt + (size_t)Cn * Bn * Lf;
    float* w2p  = env  + (size_t)Cn * Bn * Lf;
    float* hid  = w2p  + (size_t)Cn * 64 * 32 * 16;

    k_w2pad<<<(Cn * 64 * 32 * 16 + 255) / 256, 256, 0, stream>>>(conv2_w, w2p);
    k_sosfilt<<<(Cn * Bn + 127) / 128, 128, 0, stream>>>(signal, sos, filt);
    k_convs<<<Cn * Bn * NTILE_ROW, 256, 0, stream>>>(filt, conv1_w, conv1_b, conv2_b,
                                                     conv3_w, conv3_b, w2p, env);
    k_fc1<<<96, 256, 0, stream>>>(env, fc1_w, fc1_b, hid);   // 768 wave-tiles
    k_fc2<<<24, 256, 0, stream>>>(hid, fc2_w, fc2_b, out);   // 192 wave-tiles
}